// GCN_23089744183306
// MI455X (gfx1250) — compile-verified
//
#include <hip/hip_runtime.h>

typedef __attribute__((ext_vector_type(2))) float v2f;
typedef __attribute__((ext_vector_type(8))) float v8f;

#define K_DIM 128  // inner dim is 128 for both layers (d_in=128, d_hid=128)

// ---------------- degree / normalization ----------------
__global__ __launch_bounds__(256) void deg_init_kernel(float* deg, int N) {
  int i = blockIdx.x * 256 + threadIdx.x;
  if (i < N) deg[i] = 1.0f;  // self-loop contributes 1 to every target degree
}

__global__ __launch_bounds__(256) void deg_count_kernel(const int* __restrict__ col,
                                                        float* __restrict__ deg, int E) {
  int e = blockIdx.x * 256 + threadIdx.x;
  if (e < E) atomicAdd(&deg[col[e]], 1.0f);
}

__global__ __launch_bounds__(256) void dinv_kernel(float* deg, int N) {
  int i = blockIdx.x * 256 + threadIdx.x;
  if (i < N) {
    float d = deg[i];
    deg[i] = (d > 0.0f) ? rsqrtf(d) : 0.0f;  // in place: deg -> deg^{-1/2}
  }
}

// ---------------- fused GEMM + pre-scale (fp32 WMMA) ----------------
// p[r,:] = dinv[r] * (A[r,:] @ W);  acc[r,:] = p[r,:]  (self-loop seed)
// A: M x 128 row-major, W: 128 x NCOLS row-major.
// Block = 256 threads = 8 waves; each wave computes a 16-row slab, all NCOLS.
//
// W is repacked into LDS in *fragment order* so each B fragment is one aligned
// ds_load_b64: element (kstep, khalf, n, klo) at
//   sW[(kstep*2 + khalf)*STRIDE + n*2 + klo]
// STRIDE = NCOLS*2 + 32 => (STRIDE mod 64) == 32, so khalf=0 lanes use banks
// 0..31 and khalf=1 lanes use banks 32..63: conflict-free.
template <int NCOLS>
__global__ __launch_bounds__(256) void gemm_scaled_wmma(
    const float* __restrict__ A, const float* __restrict__ W,
    const float* __restrict__ dinv, float* __restrict__ p,
    float* __restrict__ acc, int M) {
  constexpr int NT = NCOLS / 16;
  constexpr int STRIDE = NCOLS * 2 + 32;   // floats per (kstep,khalf) row
  __shared__ float sW[64 * STRIDE];        // 64 = 32 ksteps * 2 halves

  for (int idx = threadIdx.x; idx < K_DIM * NCOLS; idx += 256) {
    int k = idx / NCOLS;
    int n = idx - k * NCOLS;
    int kstep = k >> 2;
    int khalfS = (k >> 1) & 1;
    int klo = k & 1;
    sW[(kstep * 2 + khalfS) * STRIDE + n * 2 + klo] = W[idx];
  }
  __syncthreads();

  const int lane = threadIdx.x & 31;
  const int wave = threadIdx.x >> 5;
  const int rowBase = blockIdx.x * 128 + wave * 16;
  const int m = lane & 15;      // A: M row within tile / B,C,D: N col within tile
  const int khalf = lane >> 4;  // 0: K=0,1 ; 1: K=2,3 (A/B f32 16x16x4 layout)

  v8f c[NT] = {};

  const int aRow = min(rowBase + m, M - 1);  // clamp loads; stores are guarded
  const float* aPtr = A + (long)aRow * K_DIM + khalf * 2;
  const float* bBase = &sW[khalf * STRIDE + m * 2];

  for (int ks = 0; ks < K_DIM / 4; ++ks) {
    v2f av = *(const v2f*)(aPtr + ks * 4);          // global_load_b64
    const float* bp = bBase + ks * 2 * STRIDE;
#pragma unroll
    for (int t = 0; t < NT; ++t) {
      v2f bv = *(const v2f*)(bp + t * 32);          // ds_load_b64, pair-aligned
      c[t] = __builtin_amdgcn_wmma_f32_16x16x4_f32(
          /*neg_a=*/false, av, /*neg_b=*/false, bv,
          /*c_mod=*/(short)0, c[t], /*reuse_a=*/false, /*reuse_b=*/false);
    }
  }

  // C/D layout: VGPR r <-> M = r (lanes 0-15), M = 8+r (lanes 16-31); N = lane&15
#pragma unroll
  for (int r = 0; r < 8; ++r) {
    int row = rowBase + r + khalf * 8;
    if (row < M) {
      float dv = dinv[row];
#pragma unroll
      for (int t = 0; t < NT; ++t) {
        float val = c[t][r] * dv;
        long o = (long)row * NCOLS + t * 16 + m;
        p[o] = val;
        acc[o] = val;
      }
    }
  }
}

// ---------------- edge scatter: acc[col] += p[row] ----------------
template <int NCOLS>
__global__ __launch_bounds__(256) void scatter_kernel(
    const float* __restrict__ p, const int* __restrict__ row,
    const int* __restrict__ col, float* __restrict__ acc, int E) {
  constexpr int CHUNKS = NCOLS / 4;  // float4 chunks per edge
  long gid = (long)blockIdx.x * 256 + threadIdx.x;
  int e = (int)(gid / CHUNKS);
  int f = (int)(gid % CHUNKS) * 4;
  if (e < E) {
    int rs = row[e];
    int ct = col[e];
    const float4 v = *(const float4*)(p + (long)rs * NCOLS + f);
    float* dst = acc + (long)ct * NCOLS + f;
    atomicAdd(dst + 0, v.x);
    atomicAdd(dst + 1, v.y);
    atomicAdd(dst + 2, v.z);
    atomicAdd(dst + 3, v.w);
  }
}

// ---------------- post-scale + bias (+ReLU) ----------------
template <int NCOLS, bool RELU>
__global__ __launch_bounds__(256) void finish_kernel(
    const float* __restrict__ acc, const float* __restrict__ dinv,
    const float* __restrict__ b, float* __restrict__ out, int M) {
  constexpr int CHUNKS = NCOLS / 4;
  long gid = (long)blockIdx.x * 256 + threadIdx.x;
  int i = (int)(gid / CHUNKS);
  int f = (int)(gid % CHUNKS) * 4;
  if (i < M) {
    float dv = dinv[i];
    float4 a = *(const float4*)(acc + (long)i * NCOLS + f);
    float4 bb = *(const float4*)(b + f);
    float4 o;
    o.x = a.x * dv + bb.x;
    o.y = a.y * dv + bb.y;
    o.z = a.z * dv + bb.z;
    o.w = a.w * dv + bb.w;
    if (RELU) {
      o.x = fmaxf(o.x, 0.0f);
      o.y = fmaxf(o.y, 0.0f);
      o.z = fmaxf(o.z, 0.0f);
      o.w = fmaxf(o.w, 0.0f);
    }
    *(float4*)(out + (long)i * NCOLS + f) = o;
  }
}

// ---------------- launcher ----------------
extern "C" void kernel_launch(void* const* d_in, const int* in_sizes, int n_in,
                              void* d_out, int out_size, void* d_ws, size_t ws_size,
                              hipStream_t stream) {
  const float* x  = (const float*)d_in[0];
  const int*   ei = (const int*)d_in[1];
  const float* W1 = (const float*)d_in[2];
  const float* b1 = (const float*)d_in[3];
  const float* W2 = (const float*)d_in[4];
  const float* b2 = (const float*)d_in[5];
  float* out = (float*)d_out;

  const int N = in_sizes[0] / 128;  // 100000
  const int E = in_sizes[1] / 2;    // 1600000
  const int* row = ei;       // edge_index[0] = sources j
  const int* col = ei + E;   // edge_index[1] = targets i

  // workspace layout (reused across layers)
  char* ws = (char*)d_ws;
  size_t szDinv = (((size_t)N * 4) + 4095) & ~(size_t)4095;
  float* dinv = (float*)ws;                        // N
  float* buf1 = (float*)(ws + szDinv);             // N*128 : p1, then h
  float* buf2 = buf1 + (size_t)N * 128;            // N*128 : acc1, then p2
  float* buf3 = buf2 + (size_t)N * 128;            // N*64  : acc2

  dim3 blk(256);
  auto cdiv = [](long a, long b) { return (unsigned)((a + b - 1) / b); };

  // normalization coefficients
  deg_init_kernel<<<cdiv(N, 256), blk, 0, stream>>>(dinv, N);
  deg_count_kernel<<<cdiv(E, 256), blk, 0, stream>>>(col, dinv, E);
  dinv_kernel<<<cdiv(N, 256), blk, 0, stream>>>(dinv, N);

  // ---- layer 1 (128 -> 128), ReLU ----
  gemm_scaled_wmma<128><<<cdiv(N, 128), blk, 0, stream>>>(x, W1, dinv, buf1, buf2, N);
  scatter_kernel<128><<<cdiv((long)E * 32, 256), blk, 0, stream>>>(buf1, row, col, buf2, E);
  finish_kernel<128, true><<<cdiv((long)N * 32, 256), blk, 0, stream>>>(buf2, dinv, b1, buf1, N);

  // ---- layer 2 (128 -> 64) ----
  gemm_scaled_wmma<64><<<cdiv(N, 128), blk, 0, stream>>>(buf1, W2, dinv, buf2, buf3, N);
  scatter_kernel<64><<<cdiv((long)E * 16, 256), blk, 0, stream>>>(buf2, row, col, buf3, E);
  finish_kernel<64, false><<<cdiv((long)N * 16, 256), blk, 0, stream>>>(buf3, dinv, b2, out, N);
}